// GTN_54838142435461
// MI455X (gfx1250) — compile-verified
//
#include <hip/hip_runtime.h>
#include <hip/hip_bf16.h>

// ---------------------------------------------------------------------------
// GTN on MI455X (gfx1250).  Heavy work = 4 channel-wise 4096^3 fp32 GEMMs
// -> V_WMMA_F32_16X16X4_F32 tiled GEMM.  Everything else is tiny.
// B tile is stored transposed in LDS so both A and B fragments are adjacent
// K-pairs -> single ds_load_b64 per fragment, no repack movs in the WMMA loop.
// ---------------------------------------------------------------------------

#define N_NODES   4096
#define N_EDGES_C 150000
#define NUM_EDGE  5
#define W_IN      512
#define W_OUT     128
#define N_TARGET  1024
#define N_CLASS   16

typedef __attribute__((ext_vector_type(2))) float v2f;
typedef __attribute__((ext_vector_type(8))) float v8f;

#define TILE_M 128
#define TILE_N 128
#define TILE_K 32

// ---------------------------------------------------------------------------
// fp32 WMMA GEMM:  C[z] = A[z] (MxK, row-major) @ B[z] (KxN, row-major)
// Optional per-column bias + ReLU epilogue.
// Block = 256 threads = 8 waves; waves arranged 4 (M) x 2 (N);
// wave tile = 32x64 = 2x4 WMMA 16x16 accumulators.
// Requires M%128==0, N%128==0, K%32==0 (true for all calls here).
// ---------------------------------------------------------------------------
__global__ __launch_bounds__(256) void gtn_gemm_wmma_f32(
    const float* __restrict__ A, const float* __restrict__ B,
    float* __restrict__ C, int M, int N, int K,
    long long sA, long long sB, long long sC,
    const float* __restrict__ bias, int relu)
{
    __shared__ float As [TILE_M][TILE_K + 4];   // 128 x 36, A tile (row-major)
    __shared__ float BsT[TILE_N][TILE_K + 4];   // 128 x 36, B tile TRANSPOSED

    const int z = blockIdx.z;
    A += (long long)z * sA;
    B += (long long)z * sB;
    C += (long long)z * sC;

    const int tileM = blockIdx.y * TILE_M;
    const int tileN = blockIdx.x * TILE_N;

    const int tid   = threadIdx.x;
    const int wave  = tid >> 5;
    const int lane  = tid & 31;
    const int waveM = (wave & 3) * 32;    // 4 waves along M
    const int waveN = (wave >> 2) * 64;   // 2 waves along N
    const int l15   = lane & 15;
    const int khalf = (lane >> 4) << 1;   // lanes 0-15 -> K+0/1, lanes 16-31 -> K+2/3

    v8f acc[2][4];
#pragma unroll
    for (int i = 0; i < 2; ++i)
#pragma unroll
        for (int j = 0; j < 4; ++j)
            acc[i][j] = (v8f){0.f, 0.f, 0.f, 0.f, 0.f, 0.f, 0.f, 0.f};

    for (int k0 = 0; k0 < K; k0 += TILE_K) {
        // Stage A tile (128x32): 1024 float4, 4 per thread, b128 LDS stores
#pragma unroll
        for (int q = tid; q < 1024; q += 256) {
            const int row = q >> 3;        // 8 float4 per row
            const int c4  = q & 7;
            const float4 v = *(const float4*)(A + (long long)(tileM + row) * K + k0 + c4 * 4);
            *(float4*)(&As[row][c4 * 4]) = v;
        }
        // Stage B tile (32x128) transposed: float4 of 4 N-values -> 4 rows of BsT
#pragma unroll
        for (int q = tid; q < 1024; q += 256) {
            const int row = q >> 5;        // k index within tile
            const int c4  = q & 31;        // n/4 index within tile
            const float4 v = *(const float4*)(B + (long long)(k0 + row) * N + tileN + c4 * 4);
            BsT[c4 * 4 + 0][row] = v.x;
            BsT[c4 * 4 + 1][row] = v.y;
            BsT[c4 * 4 + 2][row] = v.z;
            BsT[c4 * 4 + 3][row] = v.w;
        }
        __syncthreads();

#pragma unroll
        for (int kk = 0; kk < TILE_K; kk += 4) {
            const int kb = kk + khalf;
            v2f afrag[2], bfrag[4];
#pragma unroll
            for (int mf = 0; mf < 2; ++mf) {
                const int r = waveM + mf * 16 + l15;       // A: lane holds row M
                afrag[mf].x = As[r][kb];
                afrag[mf].y = As[r][kb + 1];
            }
#pragma unroll
            for (int nf = 0; nf < 4; ++nf) {
                const int cN = waveN + nf * 16 + l15;      // B: lane holds col N
                bfrag[nf].x = BsT[cN][kb];
                bfrag[nf].y = BsT[cN][kb + 1];
            }
#pragma unroll
            for (int mf = 0; mf < 2; ++mf)
#pragma unroll
                for (int nf = 0; nf < 4; ++nf)
                    acc[mf][nf] = __builtin_amdgcn_wmma_f32_16x16x4_f32(
                        false, afrag[mf], false, bfrag[nf],
                        (short)0, acc[mf][nf], false, false);
        }
        __syncthreads();
    }

    // Epilogue: C layout (ISA 7.12.2): VGPR r -> M = r (+8 for lanes 16-31), N = lane&15
#pragma unroll
    for (int mf = 0; mf < 2; ++mf) {
#pragma unroll
        for (int nf = 0; nf < 4; ++nf) {
            const int cN    = tileN + waveN + nf * 16 + l15;
            const int rBase = tileM + waveM + mf * 16 + ((lane >> 4) << 3);
            const float bv  = bias ? bias[cN] : 0.f;
#pragma unroll
            for (int r = 0; r < 8; ++r) {
                float v = acc[mf][nf][r] + bv;
                if (relu) v = v > 0.f ? v : 0.f;
                C[(long long)(rBase + r) * N + cN] = v;
            }
        }
    }
}

// ---------------------------------------------------------------------------
// Softmax of GTConv weights over the edge-type axis: F[6][5]
// rows: {l0c1 ch0, l0c1 ch1, l0c2 ch0, l0c2 ch1, l1c1 ch0, l1c1 ch1}
// ---------------------------------------------------------------------------
__global__ void gtn_softmax_w(const float* __restrict__ w1,
                              const float* __restrict__ w2,
                              const float* __restrict__ w3,
                              float* __restrict__ F)
{
    if (threadIdx.x == 0 && blockIdx.x == 0) {
        const float* ws[3] = {w1, w2, w3};
        for (int s = 0; s < 3; ++s)
            for (int c = 0; c < 2; ++c) {
                const float* r = ws[s] + c * NUM_EDGE;
                float m = r[0];
                for (int j = 1; j < NUM_EDGE; ++j) m = fmaxf(m, r[j]);
                float e[NUM_EDGE], sum = 0.f;
                for (int j = 0; j < NUM_EDGE; ++j) { e[j] = expf(r[j] - m); sum += e[j]; }
                for (int j = 0; j < NUM_EDGE; ++j) F[(s * 2 + c) * NUM_EDGE + j] = e[j] / sum;
            }
    }
}

// ---------------------------------------------------------------------------
// Scatter-add edges directly into convex-combined dense adjacencies.
// base holds nsets consecutive [n x n] matrices; set s uses coeff row fRow0+s.
// ---------------------------------------------------------------------------
__global__ __launch_bounds__(256) void gtn_scatter_adj(
    const int* __restrict__ ei, const float* __restrict__ ev,
    const float* __restrict__ F, int fRow0, int nsets,
    float* __restrict__ base, int E, int n)
{
    const long long tid   = (long long)blockIdx.x * blockDim.x + threadIdx.x;
    const long long total = (long long)NUM_EDGE * E;
    if (tid >= total) return;
    const int t = (int)(tid / E);
    const int e = (int)(tid % E);
    const int i = ei[((long long)t * 2 + 0) * E + e];
    const int j = ei[((long long)t * 2 + 1) * E + e];
    const float v = ev[(long long)t * E + e];
    const long long NN = (long long)n * n;
    for (int s = 0; s < nsets; ++s) {
        const float c = F[(fRow0 + s) * NUM_EDGE + t];
        atomicAdd(base + (long long)s * NN + (long long)i * n + j, c * v);
    }
}

// ---------------------------------------------------------------------------
// Row normalization: H[row] *= 1/sum(H[row])  (0 for empty rows).
// One block per row.
// ---------------------------------------------------------------------------
__global__ __launch_bounds__(256) void gtn_row_norm(float* __restrict__ H, int rowLen)
{
    __shared__ float red[256];
    float* p = H + (long long)blockIdx.x * rowLen;
    float s = 0.f;
    for (int i = threadIdx.x; i < rowLen; i += 256) s += p[i];
    red[threadIdx.x] = s;
    __syncthreads();
    for (int off = 128; off > 0; off >>= 1) {
        if (threadIdx.x < off) red[threadIdx.x] += red[threadIdx.x + off];
        __syncthreads();
    }
    const float d = red[0];
    const float dinv = (d == 0.f) ? 0.f : 1.f / d;
    for (int i = threadIdx.x; i < rowLen; i += 256) p[i] *= dinv;
}

// ---------------------------------------------------------------------------
// Gather target rows, concat channels, final linear -> [N_TARGET, N_CLASS]
// ---------------------------------------------------------------------------
__global__ __launch_bounds__(256) void gtn_classify(
    const float* __restrict__ Xc, const int* __restrict__ tgt,
    const float* __restrict__ lw, const float* __restrict__ lb,
    float* __restrict__ out)
{
    const int gid = blockIdx.x * blockDim.x + threadIdx.x;
    if (gid >= N_TARGET * N_CLASS) return;
    const int tg = gid >> 4;
    const int cls = gid & 15;
    const int node = tgt[tg];
    const float* x0 = Xc + (long long)node * W_OUT;
    const float* x1 = Xc + (long long)N_NODES * W_OUT + (long long)node * W_OUT;
    float s = lb[cls];
    for (int k = 0; k < W_OUT; ++k) s += x0[k] * lw[k * N_CLASS + cls];
    for (int k = 0; k < W_OUT; ++k) s += x1[k] * lw[(W_OUT + k) * N_CLASS + cls];
    out[gid] = s;
}

// ---------------------------------------------------------------------------
extern "C" void kernel_launch(void* const* d_in, const int* in_sizes, int n_in,
                              void* d_out, int out_size, void* d_ws, size_t ws_size,
                              hipStream_t stream)
{
    const int*   edge_index = (const int*)d_in[0];
    const float* edge_value = (const float*)d_in[1];
    const float* X          = (const float*)d_in[2];
    const int*   target_x   = (const int*)d_in[3];
    const float* w_l0_c1    = (const float*)d_in[4];
    const float* w_l0_c2    = (const float*)d_in[5];
    const float* w_l1_c1    = (const float*)d_in[6];
    const float* gcn_w      = (const float*)d_in[7];
    const float* gcn_b      = (const float*)d_in[8];
    const float* lin_w      = (const float*)d_in[9];
    const float* lin_b      = (const float*)d_in[10];
    float* out = (float*)d_out;
    float* ws  = (float*)d_ws;

    const int n = N_NODES, E = N_EDGES_C;
    const long long NN = (long long)n * n;

    // Workspace layout (floats):
    float* F  = ws;                                 // 30 (softmax coeffs)
    float* B0 = ws + 64;                            // [2][n][n] conv1 A, later conv3 A
    float* B1 = B0 + 2 * NN;                        // [2][n][n] conv2 A, later H1
    float* H0 = B1 + 2 * NN;                        // [2][n][n]
    float* XW = H0 + 2 * NN;                        // [n][W_OUT]
    float* Xc = XW + (long long)n * W_OUT;          // [2][n][W_OUT]

    // 1. softmax over edge types for all three GTConvs
    gtn_softmax_w<<<1, 32, 0, stream>>>(w_l0_c1, w_l0_c2, w_l1_c1, F);

    // 2. build A1 (2 ch) and A2 (2 ch) = convex-combined dense adjacencies
    hipMemsetAsync(B0, 0, (size_t)(4 * NN) * sizeof(float), stream);
    {
        const long long total = (long long)NUM_EDGE * E;
        const int blocks = (int)((total + 255) / 256);
        gtn_scatter_adj<<<blocks, 256, 0, stream>>>(edge_index, edge_value, F, 0, 4, B0, E, n);
    }

    // 3. H0 = A1 @ A2 per channel (WMMA), then row-normalize
    {
        dim3 grid(n / TILE_N, n / TILE_M, 2);
        gtn_gemm_wmma_f32<<<grid, 256, 0, stream>>>(B0, B1, H0, n, n, n, NN, NN, NN, nullptr, 0);
    }
    gtn_row_norm<<<2 * n, 256, 0, stream>>>(H0, n);

    // 4. rebuild B0 as A3 = conv(w_l1_c1, A) (2 ch)
    hipMemsetAsync(B0, 0, (size_t)(2 * NN) * sizeof(float), stream);
    {
        const long long total = (long long)NUM_EDGE * E;
        const int blocks = (int)((total + 255) / 256);
        gtn_scatter_adj<<<blocks, 256, 0, stream>>>(edge_index, edge_value, F, 4, 2, B0, E, n);
    }

    // 5. H1 = H0 @ A3 per channel (into B1), row-normalize
    {
        dim3 grid(n / TILE_N, n / TILE_M, 2);
        gtn_gemm_wmma_f32<<<grid, 256, 0, stream>>>(H0, B0, B1, n, n, n, NN, NN, NN, nullptr, 0);
    }
    gtn_row_norm<<<2 * n, 256, 0, stream>>>(B1, n);

    // 6. XW = X @ gcn_w   (4096 x 512 @ 512 x 128)
    {
        dim3 grid(W_OUT / TILE_N, n / TILE_M, 1);
        gtn_gemm_wmma_f32<<<grid, 256, 0, stream>>>(X, gcn_w, XW, n, W_OUT, W_IN, 0, 0, 0, nullptr, 0);
    }

    // 7. Xc = relu(H1 @ XW + gcn_b) per channel  (fused epilogue)
    {
        dim3 grid(W_OUT / TILE_N, n / TILE_M, 2);
        gtn_gemm_wmma_f32<<<grid, 256, 0, stream>>>(B1, XW, Xc, n, W_OUT, n,
                                                    NN, 0, (long long)n * W_OUT, gcn_b, 1);
    }

    // 8. gather targets, concat channels, final linear
    gtn_classify<<<(N_TARGET * N_CLASS + 255) / 256, 256, 0, stream>>>(Xc, target_x, lin_w, lin_b, out);
}